// SoftTruncatedGaussianMixtureAnalysis_46677704573009
// MI455X (gfx1250) — compile-verified
//
#include <hip/hip_runtime.h>
#include <hip/hip_bf16.h>
#include <math.h>

// Problem constants (from reference)
#define NN_ 16384
#define DD_ 128
#define CC_ 10
#define KK_ 16
#define DEXT_ 256              // [x, x^2] extended feature length
#define NCOL_ (CC_ * KK_)      // 160 GEMM columns
#define EPS_ 1.1920928955078125e-07f
#define LOG2PI_ 1.8378770664093454f
#define LN2_ 0.6931471805599453f

typedef float v2f __attribute__((ext_vector_type(2)));
typedef float v8f __attribute__((ext_vector_type(8)));

// ---------------- device helpers ----------------
__device__ __forceinline__ float softplusf(float x) {
    // log(1+exp(x)) = max(x,0) + log1p(exp(-|x|))   (stable, matches jax.nn.softplus)
    return fmaxf(x, 0.0f) + log1pf(expf(-fabsf(x)));
}
__device__ __forceinline__ float ndtrf(float x) {
    // Phi(x) = 0.5 * erfc(-x / sqrt(2))
    return 0.5f * erfcf(-x * 0.70710678118654752440f);
}

// ---------------- kernel 1: zero accumulators ----------------
__global__ void k_init(float* class_sums, int* counts) {
    int i = threadIdx.x;
    if (i < CC_) { class_sums[i] = 0.0f; counts[i] = 0; }
}

// ---------------- kernel 2: class histogram ----------------
__global__ void k_hist(const int* __restrict__ y, int* counts) {
    int i = blockIdx.x * blockDim.x + threadIdx.x;
    if (i < NN_) atomicAdd(&counts[y[i]], 1);
}

// ---------------- kernel 3: per-(c,k) precompute -> W fragments + cst ----------------
// W layout: fragment-ready for V_WMMA_F32_16X16X4_F32 B operand, with the two
// B-operand VGPRs interleaved per lane so the GEMM loads one b64 per step:
//   Wf[(t*64 + s)*64 + lane*2 + v]   t = col tile (=class), s = k-step,
//   v = VGPR index (0/1), lane = WMMA lane.
// B 4x16 layout mirrors the documented 32-bit A layout:
//   VGPR0: lanes0-15 -> K-row 0, lanes16-31 -> K-row 2; VGPR1: rows 1 / 3.
__device__ __forceinline__ void store_wfrag(float* Wf, int t, int nn, int dext, float val) {
    int s   = dext >> 2;
    int rem = dext & 3;
    int v   = rem & 1;
    int lane = (rem < 2) ? nn : nn + 16;
    Wf[(t * 64 + s) * 64 + lane * 2 + v] = val;
}

__global__ __launch_bounds__(128) void k_prep(const float* __restrict__ mu,
                                              const float* __restrict__ lower,
                                              const float* __restrict__ upper,
                                              const float* __restrict__ sigma,
                                              const float* __restrict__ logits_k,
                                              float* __restrict__ Wf,
                                              float* __restrict__ cst) {
    __shared__ float red[128];
    int j = blockIdx.x;            // 0 .. C*K-1
    int c = j / KK_, k = j % KK_;
    int d = threadIdx.x;           // 0 .. 127

    int base = j * DD_ + d;
    float m  = mu[base];
    float lo = lower[base];
    float up = upper[base];
    float sg = sigma[base];

    float scale = EPS_ + softplusf(sg);
    float inv   = 1.0f / scale;
    float inv2  = inv * inv;
    float z     = ndtrf((up - m) * inv) - ndtrf((lo - m) * inv);

    // --- product reduction for prod(z) ---
    red[d] = z; __syncthreads();
    for (int s = 64; s > 0; s >>= 1) { if (d < s) red[d] *= red[d + s]; __syncthreads(); }
    float log_norm = logf(red[0] + EPS_);
    __syncthreads();

    // --- sum reduction for mu^2 * inv2 ---
    red[d] = m * m * inv2; __syncthreads();
    for (int s = 64; s > 0; s >>= 1) { if (d < s) red[d] += red[d + s]; __syncthreads(); }
    float sum_m2 = red[0];
    __syncthreads();

    // --- sum reduction for log(scale) ---
    red[d] = logf(scale); __syncthreads();
    for (int s = 64; s > 0; s >>= 1) { if (d < s) red[d] += red[d + s]; __syncthreads(); }
    float sum_ls = red[0];
    __syncthreads();

    // log_softmax over K (redundant per thread; K=16, cheap)
    float mx = -INFINITY;
    for (int kk = 0; kk < KK_; ++kk) mx = fmaxf(mx, logits_k[c * KK_ + kk]);
    float se = 0.0f;
    for (int kk = 0; kk < KK_; ++kk) se += expf(logits_k[c * KK_ + kk] - mx);
    float log_cat = logits_k[c * KK_ + k] - (mx + logf(se));

    if (d == 0) {
        cst[j] = log_cat - log_norm - 0.5f * sum_m2 - sum_ls - 0.5f * (float)DD_ * LOG2PI_;
    }

    // weight rows: row d -> mu/s^2 (coef of x), row d+128 -> -0.5/s^2 (coef of x^2)
    store_wfrag(Wf, c, k, d,        m * inv2);
    store_wfrag(Wf, c, k, d + DD_, -0.5f * inv2);
}

// ---------------- kernel 4: lp_low + lp_up (soft truncation) ----------------
__global__ void k_lp(const float* __restrict__ X, const int* __restrict__ y,
                     const float* __restrict__ lower, const float* __restrict__ upper,
                     const float* __restrict__ eta_p, float* __restrict__ lp) {
    int idx = blockIdx.x * blockDim.x + threadIdx.x;
    if (idx >= NN_ * KK_) return;
    float eta = eta_p[0];
    if (eta == 0.0f) {                      // softplus(0) = ln2, summed twice over D
        lp[idx] = -2.0f * (float)DD_ * LN2_;
        return;
    }
    int n = idx / KK_, k = idx % KK_;
    int c = y[n];
    const float* xr = X + (size_t)n * DD_;
    const float* lo = lower + ((size_t)c * KK_ + k) * DD_;
    const float* up = upper + ((size_t)c * KK_ + k) * DD_;
    float s = 0.0f;
    for (int d = 0; d < DD_; ++d) {
        float x = xr[d];
        s += softplusf(-eta * (x - lo[d])) + softplusf(-eta * (up[d] - x));
    }
    lp[idx] = -s;
}

// ---------------- kernel 5: WMMA GEMM + fused epilogue ----------------
// Block: 320 threads = 10 waves. Wave t owns column tile t (class t, cols t*16..t*16+15).
// Block handles one 16-row tile of X, staged (with x^2 halves) in LDS.
#define XT_STRIDE 260   // 260 % 64 = 4 -> 16 rows hit 16 distinct banks
__global__ __launch_bounds__(320) void k_gemm(const float* __restrict__ X,
                                              const int* __restrict__ y,
                                              const float* __restrict__ resp,
                                              const float* __restrict__ Wf,
                                              const float* __restrict__ cst,
                                              const float* __restrict__ lp,
                                              float* __restrict__ class_sums) {
    __shared__ float xt[16 * XT_STRIDE];
    __shared__ int   ytile[16];
    int n0   = blockIdx.x * 16;
    int tid  = threadIdx.x;
    int lane = tid & 31;
    int t    = tid >> 5;                 // wave id == column tile == class

    if (tid < 16) ytile[tid] = y[n0 + tid];

    // Stage extended features [x, x^2] for 16 rows into LDS (coalesced).
    for (int idx = tid; idx < 16 * DEXT_; idx += 320) {
        int r  = idx >> 8;
        int dd = idx & 255;
        float v = X[(size_t)(n0 + r) * DD_ + (dd & (DD_ - 1))];
        if (dd >= DD_) v *= v;
        xt[r * XT_STRIDE + dd] = v;
    }
    __syncthreads();

    // A fragment addressing per ISA 32-bit A 16x4 layout:
    //   lanes 0-15: VGPR0=K0, VGPR1=K1 ; lanes 16-31: VGPR0=K2, VGPR1=K3 ; M = lane%16
    int mrow  = lane & 15;
    int khalf = (lane < 16) ? 0 : 2;
    const float* wbase = Wf + (size_t)t * 64 * 64 + lane * 2;

    v8f acc = {};
    for (int s = 0; s < 64; ++s) {
        int k0 = s * 4;
        v2f a;
        a[0] = xt[mrow * XT_STRIDE + k0 + khalf];
        a[1] = xt[mrow * XT_STRIDE + k0 + khalf + 1];
        v2f b = *reinterpret_cast<const v2f*>(wbase + s * 64);   // one b64 per lane
        // D = A(16x4,f32) * B(4x16,f32) + C
        acc = __builtin_amdgcn_wmma_f32_16x16x4_f32(false, a, false, b,
                                                    (short)0, acc, false, false);
    }

    // Epilogue: D layout -> VGPR i holds (M=i | M=i+8), N = lane%16.
    int nn    = lane & 15;
    int mbase = (lane < 16) ? 0 : 8;
    float cstv = cst[t * KK_ + nn];
    float partial = 0.0f;
#pragma unroll
    for (int i = 0; i < 8; ++i) {
        int m = mbase + i;
        int n = n0 + m;
        if (ytile[m] == t) {
            float logpdf = cstv + acc[i] + lp[n * KK_ + nn];
            partial += resp[((size_t)t * NN_ + n) * KK_ + nn] * logpdf;
        }
    }
    // wave32 reduction
    for (int off = 16; off > 0; off >>= 1)
        partial += __shfl_xor(partial, off, 32);
    if (lane == 0) atomicAdd(&class_sums[t], partial);
}

// ---------------- kernel 6: finalize ----------------
__global__ void k_final(const float* __restrict__ class_sums,
                        const int* __restrict__ counts, float* out) {
    if (blockIdx.x == 0 && threadIdx.x == 0) {
        float loss = 0.0f;
        for (int c = 0; c < CC_; ++c)
            loss += class_sums[c] / ((float)counts[c] * (float)KK_);
        out[0] = -loss;
    }
}

// ---------------- launcher ----------------
extern "C" void kernel_launch(void* const* d_in, const int* in_sizes, int n_in,
                              void* d_out, int out_size, void* d_ws, size_t ws_size,
                              hipStream_t stream) {
    const float* X      = (const float*)d_in[0];
    const int*   y      = (const int*)  d_in[1];
    const float* resp   = (const float*)d_in[2];
    const float* mu     = (const float*)d_in[3];
    const float* lower  = (const float*)d_in[4];
    const float* upper  = (const float*)d_in[5];
    const float* sigma  = (const float*)d_in[6];
    const float* logits = (const float*)d_in[7];
    const float* eta    = (const float*)d_in[8];
    float* out = (float*)d_out;

    // workspace layout (floats)
    float* ws       = (float*)d_ws;
    float* Wf       = ws;                                   // 10*64*64      = 40960 f
    float* cst      = ws + 40960;                           // 160 f
    float* csums    = ws + 40960 + 160;                     // 10 f
    int*   counts   = (int*)(ws + 40960 + 160 + 16);        // 10 i (padded)
    float* lp       = ws + 41216;                           // N*K = 262144 f

    k_init<<<1, 32, 0, stream>>>(csums, counts);
    k_hist<<<(NN_ + 255) / 256, 256, 0, stream>>>(y, counts);
    k_prep<<<NCOL_, 128, 0, stream>>>(mu, lower, upper, sigma, logits, Wf, cst);
    k_lp<<<(NN_ * KK_ + 255) / 256, 256, 0, stream>>>(X, y, lower, upper, eta, lp);
    k_gemm<<<NN_ / 16, 320, 0, stream>>>(X, y, resp, Wf, cst, lp, csums);
    k_final<<<1, 1, 0, stream>>>(csums, counts, out);
}